// simple_feature_renet_72739566125504
// MI455X (gfx1250) — compile-verified
//
#include <hip/hip_runtime.h>

typedef __attribute__((ext_vector_type(2))) float v2f;
typedef __attribute__((ext_vector_type(8))) float v8f;

// ---------------- padded weight panels in d_ws (float offsets) ----------------
// Each panel is [Npad][Kstride], zero-filled for k>=K (required: ghost-k A reads
// are garbage and must be annihilated by B==0) and n>=N.
#define WOFF_G1   0        // [176][128] l01 | r01   (K=127)
#define WOFF_G2   22528    // [32][16]   cb1         (K=15)
#define WOFF_G3   23040    // [80][52]   c1          (K=50)
#define WOFF_G4   27200    // [32][84]   l1          (K=81)
#define WOFF_G5   29888    // [32][84]   ra          (K=81)
#define WOFF_G6   32576    // [128][152] c2          (K=150)
#define WOFF_G7   52032    // [128][76]  ca          (K=75)
#define WOFF_G8   61760    // [32][252]  ca2         (K=250)
#define WOFF_G9   69824    // [16][32]   l02         (K=30)
#define WOFF_G10  70336    // [16][32]   r02         (K=30)
#define WS_FLOATS 70848    // 283,392 bytes of workspace

// ---------------- LDS plan (float offsets), with liveness overlays ------------
// Region B (0..10336): early {xs,u,S,t2pad} then t4p          (dead-by-G4 / from-G4)
// Region A (21844..47380): t3p (G3..G4) then {x2p,qp,spb,rp,zc} (from G6)
#define L_XS     0        // [5][128]                 640   E0..G1
#define L_U      640      // [5][84]                  420   G1..E1
#define L_S      1060     // [96]                      96   E0..E1
#define L_T2     1160     // [5][8][84] halo rows    3360   E1..G2
#define L_T4P    0        // [304][34]              10336   G4..G6
#define L_V      10336    // [5][84]                  420   G1..G2
#define L_X1P    10756    // [132][84]              11088   G2..G5
#define L_T3P    21844    // [304][84]              25536   G3..G4
#define L_X2P    21844    // [384][34]              13056   G6..G10
#define L_QP     34900    // [64][34]                2176   G8..G9
#define L_SPB    37076    // [52][18]                 936   G9..E2
#define L_RP     38012    // [376][18]               6768   G10..E2
#define L_ZC     44780    // [16]                      16   E2
#define L_PP     47380    // [132][34]               4488   G5..G7
#define LDS_TOTAL 51868   // 207,472 bytes (< 320 KB/WG)

struct Params {
    const float *x;
    const float *ca1_w, *ca1_b, *cb1_b, *cc1_w, *cc1_b, *r01_b, *l01_w, *l01_b;
    const float *c1_b,  *l1_b,  *c2_b,  *ca_b,  *ra_b;
    const float *ca2_b, *l02_b, *cb2_w, *cb2_b, *cc2_w, *cc2_b, *r02_b;
    const float *fl_w,  *fl_b;
    const float *ws;                 // padded weight panels
    float *out;
};

// Guard-free implicit-GEMM tile engine on V_WMMA_F32_16X16X4_F32.
// loadA2(m,ka)/loadB2(n,ka) return the lane's 2-element fragment unconditionally
// (panels are padded so ghost reads stay in-bounds; ghost-k weights are zero).
// k unrolled by 2 so both fragment pairs issue before the first WMMA.
template <class LA2, class LB2, class ST>
__device__ __forceinline__ void wmma_gemm(int M, int N, int Ksteps,
                                          LA2 loadA2, LB2 loadB2, ST store,
                                          int lane, int wave, int nwaves)
{
    const int half = lane >> 4;      // fragment half: k pair select
    const int lm   = lane & 15;
    const int mt = (M + 15) >> 4, nt = (N + 15) >> 4;
    for (int t = wave; t < mt * nt; t += nwaves) {
        const int tm  = t / nt, tn = t % nt;
        const int row = tm * 16 + lm;
        const int col = tn * 16 + lm;
        v8f acc = {0.f, 0.f, 0.f, 0.f, 0.f, 0.f, 0.f, 0.f};
        int s = 0;
        for (; s + 2 <= Ksteps; s += 2) {
            const int ka0 = s * 4 + 2 * half;
            v2f a0 = loadA2(row, ka0),     b0 = loadB2(col, ka0);
            v2f a1 = loadA2(row, ka0 + 4), b1 = loadB2(col, ka0 + 4);
            acc = __builtin_amdgcn_wmma_f32_16x16x4_f32(false, a0, false, b0, (short)0, acc, false, false);
            acc = __builtin_amdgcn_wmma_f32_16x16x4_f32(false, a1, false, b1, (short)0, acc, false, false);
        }
        if (s < Ksteps) {
            const int ka = s * 4 + 2 * half;
            v2f a = loadA2(row, ka), b = loadB2(col, ka);
            acc = __builtin_amdgcn_wmma_f32_16x16x4_f32(false, a, false, b, (short)0, acc, false, false);
        }
#pragma unroll
        for (int i = 0; i < 8; ++i) {
            const int m = tm * 16 + half * 8 + i;   // D: VGPR i -> M = i + 8*half
            const int n = tn * 16 + lm;
            if (m < M && n < N) store(m, n, acc[i]);
        }
    }
}

__global__ void __launch_bounds__(256)
feature_renet_fused_wmma(Params p)
{
    extern __shared__ float sm[];
    const int tid  = threadIdx.x;
    const int lane = tid & 31;
    const int wave = tid >> 5;
    const int NW   = 8;
    const int b    = blockIdx.x;

    float* xs  = sm + L_XS;
    float* u   = sm + L_U;
    float* Ss  = sm + L_S;
    float* t2  = sm + L_T2;
    float* t4p = sm + L_T4P;
    float* v   = sm + L_V;
    float* x1p = sm + L_X1P;
    float* t3p = sm + L_T3P;
    float* x2p = sm + L_X2P;
    float* qp  = sm + L_QP;
    float* spb = sm + L_SPB;
    float* rp  = sm + L_RP;
    float* zc  = sm + L_ZC;
    float* pp  = sm + L_PP;
    const float* ws = p.ws;

    // ---- E0: stage x (pad w to 128 with zeros); S[o] = sum_w l01_w[o,w]
    for (int i = tid; i < 5 * 128; i += 256) {
        const int h = i >> 7, w = i & 127;
        xs[i] = (w < 127) ? p.x[(size_t)b * 635 + h * 127 + w] : 0.f;
    }
    for (int o = tid; o < 81; o += 256) {
        float s = 0.f;
        for (int w = 0; w < 127; ++w) s += p.l01_w[o * 127 + w];
        Ss[o] = s;
    }
    __syncthreads();

    // ---- G1: [u | v] = x * [l01 | r01]^T  (M=5, N=162, K=127, Ksteps=32)
    wmma_gemm(5, 162, 32,
        [&](int m, int ka) { return *(const v2f*)(xs + m * 128 + ka); },
        [&](int n, int ka) { return *(const v2f*)(ws + WOFF_G1 + n * 128 + ka); },
        [&](int m, int n, float val) {
            if (n < 81) u[m * 84 + n] = val; else v[m * 84 + (n - 81)] = val;
        }, lane, wave, NW);
    __syncthreads();

    // ---- E1: t2 panel [c][hh+1][w] with zero halo rows (hh=-1,5) and zero pad cols.
    //      t2[c,h,o] = relu(3-tap(u) + ca1_b[c]*S[o] + l01_b[o])  (conv(ca1)+lin(l01) fused)
    for (int i = tid; i < 5 * 8 * 84; i += 256) {
        const int c = i / 672, r = i % 672, hh1 = r / 84, w = r % 84;
        float val = 0.f;
        if (hh1 >= 1 && hh1 <= 5 && w < 81) {
            const int h = hh1 - 1;
            float acc = p.ca1_b[c] * Ss[w] + p.l01_b[w];
            if (h > 0) acc += p.ca1_w[c * 3 + 0] * u[(h - 1) * 84 + w];
            acc += p.ca1_w[c * 3 + 1] * u[h * 84 + w];
            if (h < 4) acc += p.ca1_w[c * 3 + 2] * u[(h + 1) * 84 + w];
            val = fmaxf(acc, 0.f);
        }
        t2[i] = val;
    }
    __syncthreads();

    // ---- G2: conv cb1 (pad 1, halo rows in panel) + residual cc1∘(lin r01) -> x1
    //      M=(h,w)=405, N=25, K=(c,kh)=15, Ksteps=4
    wmma_gemm(405, 25, 4,
        [&](int m, int ka) {
            const int h = m / 81, w = m % 81;
            const int c0 = ka / 3,      r0 = ka - c0 * 3;
            const int c1 = (ka + 1) / 3, r1 = (ka + 1) - c1 * 3;
            v2f r;
            r.x = t2[c0 * 672 + (h + r0) * 84 + w];
            r.y = t2[c1 * 672 + (h + r1) * 84 + w];
            return r;
        },
        [&](int n, int ka) { return *(const v2f*)(ws + WOFF_G2 + n * 16 + ka); },
        [&](int m, int n, float val) {
            const int h = m / 81, w = m % 81;
            x1p[(n * 5 + h) * 84 + w] =
                val + p.cb1_b[n] + p.cc1_b[n] + p.cc1_w[n] * (v[h * 84 + w] + p.r01_b[w]);
        }, lane, wave, NW);
    __syncthreads();

    // ---- G3: conv c1 (k=2): M=(h,w)=324, N=75, K=(ci,kh)=50, Ksteps=13
    wmma_gemm(324, 75, 13,
        [&](int m, int ka) {
            const int h = m / 81, w = m % 81;
            const float* base = x1p + ((ka >> 1) * 5 + h) * 84 + w;  // ka even: kh pair 0,1
            v2f r; r.x = base[0]; r.y = base[84]; return r;
        },
        [&](int n, int ka) { return *(const v2f*)(ws + WOFF_G3 + n * 52 + ka); },
        [&](int m, int n, float val) {
            const int h = m / 81, w = m % 81;
            t3p[(n * 4 + h) * 84 + w] = val + p.c1_b[n];
        }, lane, wave, NW);
    __syncthreads();

    // ---- G4: t4 = relu(lin l1 of t3)  (M=300, K=81, N=30, Ksteps=21)
    wmma_gemm(300, 30, 21,
        [&](int m, int ka) { return *(const v2f*)(t3p + m * 84 + ka); },
        [&](int n, int ka) { return *(const v2f*)(ws + WOFF_G4 + n * 84 + ka); },
        [&](int m, int n, float val) { t4p[m * 34 + n] = fmaxf(val + p.l1_b[n], 0.f); },
        lane, wave, NW);
    // ---- G5 (independent): pp = lin ra of x1  (M=125, K=81, N=30)
    wmma_gemm(125, 30, 21,
        [&](int m, int ka) { return *(const v2f*)(x1p + m * 84 + ka); },
        [&](int n, int ka) { return *(const v2f*)(ws + WOFF_G5 + n * 84 + ka); },
        [&](int m, int n, float val) { pp[m * 34 + n] = val + p.ra_b[n]; },
        lane, wave, NW);
    __syncthreads();

    // ---- G6: conv c2 (k=2) -> x2  (M=(h,w)=90, N=125, K=150, Ksteps=38)
    wmma_gemm(90, 125, 38,
        [&](int m, int ka) {
            const int h = m / 30, w = m % 30;
            const float* base = t4p + ((ka >> 1) * 4 + h) * 34 + w;
            v2f r; r.x = base[0]; r.y = base[34]; return r;
        },
        [&](int n, int ka) { return *(const v2f*)(ws + WOFF_G6 + n * 152 + ka); },
        [&](int m, int n, float val) {
            const int h = m / 30, w = m % 30;
            x2p[(n * 3 + h) * 34 + w] = val + p.c2_b[n];
        }, lane, wave, NW);
    __syncthreads();

    // ---- G7: conv ca (k=3) residual, += into x2  (M=90, N=125, K=75, Ksteps=19)
    wmma_gemm(90, 125, 19,
        [&](int m, int ka) {
            const int h = m / 30, w = m % 30;
            const int c0 = ka / 3,       r0 = ka - c0 * 3;
            const int c1 = (ka + 1) / 3, r1 = (ka + 1) - c1 * 3;
            v2f r;
            r.x = pp[(c0 * 5 + h + r0) * 34 + w];
            r.y = pp[(c1 * 5 + h + r1) * 34 + w];
            return r;
        },
        [&](int n, int ka) { return *(const v2f*)(ws + WOFF_G7 + n * 76 + ka); },
        [&](int m, int n, float val) {
            const int h = m / 30, w = m % 30;
            x2p[(n * 3 + h) * 34 + w] += val + p.ca_b[n];
        }, lane, wave, NW);
    __syncthreads();

    // ---- G8: conv ca2 (k=2) -> q  (M=(h,w)=60, N=25, K=250, Ksteps=63)
    wmma_gemm(60, 25, 63,
        [&](int m, int ka) {
            const int h = m / 30, w = m % 30;
            const float* base = x2p + ((ka >> 1) * 3 + h) * 34 + w;
            v2f r; r.x = base[0]; r.y = base[34]; return r;
        },
        [&](int n, int ka) { return *(const v2f*)(ws + WOFF_G8 + n * 252 + ka); },
        [&](int m, int n, float val) {
            const int h = m / 30, w = m % 30;
            qp[(n * 2 + h) * 34 + w] = val + p.ca2_b[n];
        }, lane, wave, NW);
    // ---- G10 (independent): r = lin r02 of x2  (M=375, N=10, K=30, Ksteps=8)
    wmma_gemm(375, 10, 8,
        [&](int m, int ka) { return *(const v2f*)(x2p + m * 34 + ka); },
        [&](int n, int ka) { return *(const v2f*)(ws + WOFF_G10 + n * 32 + ka); },
        [&](int m, int n, float val) { rp[m * 18 + n] = val + p.r02_b[n]; },
        lane, wave, NW);
    __syncthreads();

    // ---- G9: s = relu(lin l02 of q)  (M=50, N=10, K=30, Ksteps=8)
    wmma_gemm(50, 10, 8,
        [&](int m, int ka) { return *(const v2f*)(qp + m * 34 + ka); },
        [&](int n, int ka) { return *(const v2f*)(ws + WOFF_G9 + n * 32 + ka); },
        [&](int m, int n, float val) { spb[m * 18 + n] = fmaxf(val + p.l02_b[n], 0.f); },
        lane, wave, NW);
    __syncthreads();

    // ---- E2: N=1 tails on VALU: cb2, cc2, broadcast add, final lin
    for (int w = tid; w < 10; w += 256) {
        float acc = p.cb2_b[0];
        for (int k = 0; k < 50; ++k) acc += p.cb2_w[k] * spb[k * 18 + w];  // rows == (ci*2+kh)
        zc[w] = acc;
    }
    __syncthreads();
    for (int h = tid; h < 3; h += 256) {
        float o = p.fl_b[0];
        for (int w = 0; w < 10; ++w) {
            float y = p.cc2_b[0];
            for (int ci = 0; ci < 125; ++ci) y += p.cc2_w[ci] * rp[(ci * 3 + h) * 18 + w];
            o += p.fl_w[w] * (zc[w] + y);
        }
        p.out[(size_t)b * 3 + h] = o;
    }
}

// ---------------- weight padding prep kernels ----------------
__global__ void pad_w(float* dst, const float* src, int N, int K, int Ks, int total)
{
    const int i = blockIdx.x * 256 + threadIdx.x;
    if (i >= total) return;
    const int n = i / Ks, k = i - n * Ks;
    dst[i] = (n < N && k < K) ? src[n * K + k] : 0.f;
}
__global__ void pad_w_cat(float* dst, const float* s0, const float* s1,
                          int Nh, int K, int Ks, int total)
{
    const int i = blockIdx.x * 256 + threadIdx.x;
    if (i >= total) return;
    const int n = i / Ks, k = i - n * Ks;
    float val = 0.f;
    if (k < K) {
        if (n < Nh) val = s0[n * K + k];
        else if (n < 2 * Nh) val = s1[(n - Nh) * K + k];
    }
    dst[i] = val;
}

extern "C" void kernel_launch(void* const* d_in, const int* in_sizes, int n_in,
                              void* d_out, int out_size, void* d_ws, size_t ws_size,
                              hipStream_t stream)
{
    (void)n_in; (void)out_size; (void)ws_size;
    const float* in[33];
    for (int i = 0; i < 33; ++i) in[i] = (const float*)d_in[i];
    float* ws = (float*)d_ws;

    // Stage zero-padded weight panels into workspace (deterministic, every call).
    auto pad = [&](int woff, const float* src, int N, int K, int Ks, int Npad) {
        const int tot = Npad * Ks;
        pad_w<<<(tot + 255) / 256, 256, 0, stream>>>(ws + woff, src, N, K, Ks, tot);
    };
    {
        const int tot = 176 * 128;
        pad_w_cat<<<(tot + 255) / 256, 256, 0, stream>>>(ws + WOFF_G1, in[3], in[9],
                                                         81, 127, 128, tot);
    }
    pad(WOFF_G2,  in[5],  25, 15,  16,  32);   // cb1_w
    pad(WOFF_G3,  in[11], 75, 50,  52,  80);   // c1_w
    pad(WOFF_G4,  in[13], 30, 81,  84,  32);   // l1_w
    pad(WOFF_G5,  in[19], 30, 81,  84,  32);   // ra_w
    pad(WOFF_G6,  in[15], 125, 150, 152, 128); // c2_w
    pad(WOFF_G7,  in[17], 125, 75,  76,  128); // ca_w
    pad(WOFF_G8,  in[21], 25, 250, 252, 32);   // ca2_w
    pad(WOFF_G9,  in[23], 10, 30,  32,  16);   // l02_w
    pad(WOFF_G10, in[29], 10, 30,  32,  16);   // r02_w

    Params p;
    p.x     = in[0];
    p.ca1_w = in[1];  p.ca1_b = in[2];
    p.l01_w = in[3];  p.l01_b = in[4];
    p.cb1_b = in[6];  p.cc1_w = in[7];  p.cc1_b = in[8];  p.r01_b = in[10];
    p.c1_b  = in[12]; p.l1_b  = in[14]; p.c2_b  = in[16];
    p.ca_b  = in[18]; p.ra_b  = in[20];
    p.ca2_b = in[22]; p.l02_b = in[24];
    p.cb2_w = in[25]; p.cb2_b = in[26];
    p.cc2_w = in[27]; p.cc2_b = in[28]; p.r02_b = in[30];
    p.fl_w  = in[31]; p.fl_b  = in[32];
    p.ws    = ws;
    p.out   = (float*)d_out;

    const int B = in_sizes[0] / 635;                          // x is [B,1,5,127]
    const size_t smem = (size_t)LDS_TOTAL * sizeof(float);    // ~203 KB
    (void)hipFuncSetAttribute((const void*)feature_renet_fused_wmma,
                              hipFuncAttributeMaxDynamicSharedMemorySize, (int)smem);
    feature_renet_fused_wmma<<<B, 256, smem, stream>>>(p);
}